// GCN_22119081574524
// MI455X (gfx1250) — compile-verified
//
#include <hip/hip_runtime.h>

typedef __attribute__((ext_vector_type(2))) float v2f;
typedef __attribute__((ext_vector_type(8))) float v8f;

constexpr int D_IN  = 64;
constexpr int D_HID = 64;
constexpr int N_CLS = 16;

// ---------------------------------------------------------------------------
// Kernel 1: per-node fused MLP   y[v] = relu(x[v] @ W1 + b1) @ W2 + b2
// using exact-f32 matrix cores (V_WMMA_F32_16X16X4_F32, K chained 16x).
// One wave32 handles a 16-node tile. 8 waves per 256-thread block.
// ---------------------------------------------------------------------------
__global__ void __launch_bounds__(256) node_mlp_wmma(
    const float* __restrict__ x, const float* __restrict__ W1,
    const float* __restrict__ b1, const float* __restrict__ W2,
    const float* __restrict__ b2, float* __restrict__ y,
    int n_nodes, int n_tiles) {
  __shared__ float hS[8][16 * D_HID];   // 8 waves x (16x64 f32) = 32 KB

  const int lane   = threadIdx.x & 31;
  const int wv     = threadIdx.x >> 5;
  const int tile   = blockIdx.x * 8 + wv;
  const bool active = tile < n_tiles;

  const int m      = lane & 15;   // A-frag row / B,C,D-frag column (lane&15)
  const int half   = lane >> 4;   // lanes 16-31 are the "upper" half
  const int koff   = half * 2;    // A/B: upper half holds K+2,K+3
  const int rowoff = half * 8;    // C/D: upper half holds rows 8..15
  float* hw = hS[wv];

  v2f aX[16];
  if (active) {
    const int node0 = tile * 16;
    int xrow = node0 + m;
    if (xrow >= n_nodes) xrow = n_nodes - 1;     // clamp (tail-safe)
    const float* xr = x + (size_t)xrow * D_IN;
    #pragma unroll
    for (int k = 0; k < 16; ++k)                 // whole 16x64 A tile: 16 x b64 loads
      aX[k] = *(const v2f*)(xr + 4 * k + koff);

    // ----- layer 1: h(16x64) = relu(X(16x64) @ W1(64x64) + b1), 4 N-tiles
    #pragma unroll
    for (int nt = 0; nt < 4; ++nt) {
      const int ncol = nt * 16 + m;
      const float bb = b1[ncol];
      v8f c = {bb, bb, bb, bb, bb, bb, bb, bb};  // bias folded into accumulator
      #pragma unroll
      for (int k = 0; k < 16; ++k) {
        const int kk = 4 * k + koff;
        v2f b;
        b.x = W1[(kk + 0) * D_HID + ncol];
        b.y = W1[(kk + 1) * D_HID + ncol];
        c = __builtin_amdgcn_wmma_f32_16x16x4_f32(false, aX[k], false, b,
                                                  (short)0, c, false, false);
      }
      #pragma unroll
      for (int r = 0; r < 8; ++r) {              // ReLU on accumulator -> LDS
        float v = c[r];
        hw[(r + rowoff) * D_HID + ncol] = v > 0.f ? v : 0.f;
      }
    }
  }
  __syncthreads();   // re-fragment h: D-layout (col-per-lane) -> A-layout (row-per-lane)
  if (active) {
    const int node0 = tile * 16;
    // ----- layer 2: y(16x16) = h(16x64) @ W2(64x16) + b2
    const float bb = b2[m];
    v8f c = {bb, bb, bb, bb, bb, bb, bb, bb};
    #pragma unroll
    for (int k = 0; k < 16; ++k) {
      const int kk = 4 * k + koff;
      v2f a, b;
      a.x = hw[m * D_HID + kk + 0];
      a.y = hw[m * D_HID + kk + 1];
      b.x = W2[(kk + 0) * N_CLS + m];
      b.y = W2[(kk + 1) * N_CLS + m];
      c = __builtin_amdgcn_wmma_f32_16x16x4_f32(false, a, false, b,
                                                (short)0, c, false, false);
    }
    #pragma unroll
    for (int r = 0; r < 8; ++r) {
      const int row = node0 + r + rowoff;
      if (row < n_nodes) y[(size_t)row * N_CLS + m] = c[r];
    }
  }
}

// ---------------------------------------------------------------------------
// Kernel 2: out[e] = y[col[col[e]]]   (double int64 gather; y lives in L2)
// 4 lanes per edge, each moves one float4 -> contiguous 64B coalesced writes.
// ---------------------------------------------------------------------------
__global__ void __launch_bounds__(256) edge_gather(
    const long long* __restrict__ col, const float* __restrict__ y,
    float* __restrict__ out, int n_edges) {
  const int t = blockIdx.x * blockDim.x + threadIdx.x;
  const int e = t >> 2;
  const int q = t & 3;
  if (e < n_edges) {
    const long long c1 = col[e];        // node id < n_nodes <= n_edges
    const long long c2 = col[c1];       // reference's h[col] re-index collapsed
    const float4* src = (const float4*)(y + (size_t)c2 * N_CLS);
    float4* dst = (float4*)(out + (size_t)e * N_CLS);
    dst[q] = src[q];
  }
}

extern "C" void kernel_launch(void* const* d_in, const int* in_sizes, int n_in,
                              void* d_out, int out_size, void* d_ws, size_t ws_size,
                              hipStream_t stream) {
  const float*     x  = (const float*)d_in[0];
  const float*     W1 = (const float*)d_in[1];
  const float*     b1 = (const float*)d_in[2];
  const float*     W2 = (const float*)d_in[3];
  const float*     b2 = (const float*)d_in[4];
  const long long* ei = (const long long*)d_in[5];   // edge_index is int64 [2,E]

  const int n_nodes = in_sizes[0] / D_IN;
  const int n_edges = in_sizes[5] / 2;
  const long long* col = ei + n_edges;               // second row: edge_index[1]

  float* y = (float*)d_ws;                           // n_nodes * 16 f32 (3.2 MB)

  const int n_tiles = (n_nodes + 15) / 16;
  dim3 blk(256);
  dim3 g1((unsigned)((n_tiles + 7) / 8));
  node_mlp_wmma<<<g1, blk, 0, stream>>>(x, W1, b1, W2, b2, y, n_nodes, n_tiles);

  const long long tthreads = (long long)n_edges * 4;
  dim3 g2((unsigned)((tthreads + 255) / 256));
  edge_gather<<<g2, blk, 0, stream>>>(col, y, (float*)d_out, n_edges);
}